// VectorQuantize_78898549227947
// MI455X (gfx1250) — compile-verified
//
#include <hip/hip_runtime.h>

// VQ-VAE quantize for MI455X (gfx1250, wave32).
// Dominant cost: 32768x8192x256 f32 GEMM (137.4 GFLOP) -> V_WMMA_F32_16X16X4_F32.
// Codebook (8MB) is L2-resident; dist kernel double-buffers 32x128 K-chunk
// tiles through LDS with GLOBAL_LOAD_ASYNC_TO_LDS_B128 (ASYNCcnt-tracked) so
// staging overlaps the WMMA stream.

typedef __attribute__((ext_vector_type(2))) float v2f;
typedef __attribute__((ext_vector_type(8))) float v8f;

#define WMMA_F32(a, b, c) \
  __builtin_amdgcn_wmma_f32_16x16x4_f32(false, (a), false, (b), (short)0, (c), false, false)

constexpr int Dk     = 256;       // codebook dim
constexpr int NCB    = 8192;      // codebook size
constexpr int MROWS  = 16 * 2048; // B*T = 32768
constexpr int KSTEPS = Dk / 4;    // 64 WMMA k-steps (K=4 each)

// ---------------------------------------------------------------------------
// codebook[n][d] = sum_k cw[n][k]*pw[d][k] + pb[d]   (NT gemm, one wave/tile)
// ---------------------------------------------------------------------------
__global__ __launch_bounds__(128, 1)
void vq_proj_kernel(const float* __restrict__ cw, const float* __restrict__ pw,
                    const float* __restrict__ pb, float* __restrict__ cb) {
  const int tid = threadIdx.x;
  const int l = tid & 31, w = tid >> 5;
  const int gw = blockIdx.x * 4 + w;       // global wave id, 8192 tiles total
  const int m0 = (gw >> 4) * 16;           // codebook row tile
  const int d0 = (gw & 15) * 16;           // output-dim tile
  const int ln = l & 15, h = l >> 4;

  const float* arow = cw + (size_t)(m0 + ln) * Dk + 2 * h;
  const float* brow = pw + (size_t)(d0 + ln) * Dk + 2 * h;

  v8f acc = {};
#pragma unroll
  for (int s = 0; s < KSTEPS; ++s) {
    v2f a = *(const v2f*)(arow + 4 * s);
    v2f b = *(const v2f*)(brow + 4 * s);
    acc = WMMA_F32(a, b, acc);
  }
  const float bias = pb[d0 + ln];
#pragma unroll
  for (int r = 0; r < 8; ++r) {
    // C layout: vgpr r = row r (lanes 0-15) / row r+8 (lanes 16-31), col = ln
    cb[(size_t)(m0 + r + 8 * h) * Dk + d0 + ln] = acc[r] + bias;
  }
}

// ---------------------------------------------------------------------------
// c2[n] = ||codebook[n]||^2   (one wave per row)
// ---------------------------------------------------------------------------
__global__ __launch_bounds__(256, 1)
void vq_c2_kernel(const float* __restrict__ cb, float* __restrict__ c2) {
  const int tid = threadIdx.x;
  const int l = tid & 31, w = tid >> 5;
  const int row = blockIdx.x * 8 + w;
  const float4* p = (const float4*)(cb + (size_t)row * Dk) + l;
  float4 x = p[0];
  float4 y = p[32];
  float s = x.x * x.x + x.y * x.y + x.z * x.z + x.w * x.w +
            y.x * y.x + y.y * y.y + y.z * y.z + y.w * y.w;
#pragma unroll
  for (int off = 16; off >= 1; off >>= 1) s += __shfl_xor(s, off, 32);
  if (l == 0) c2[row] = s;
}

// ---------------------------------------------------------------------------
// Distance + argmin. Block = 8 waves = 64 rows x all 8192 codes.
// A (16x256 f32 per wave) lives in 128 VGPRs. Codebook streamed as 512 units
// (32 codes x 128-float K-chunk each) through a double-buffered LDS tile via
// async global->LDS copies; s_wait_asynccnt + barrier pipelines stage/compute.
// ---------------------------------------------------------------------------
__global__ __launch_bounds__(256, 1)
void vq_dist_kernel(const float* __restrict__ enc, const float* __restrict__ cb,
                    const float* __restrict__ c2g, int* __restrict__ outidx) {
  __shared__ float buf[2 * 32 * 132];   // 2 x (32 rows x 128 floats + 4 pad)
  __shared__ float redv[64][2];
  __shared__ int   redi[64][2];

  const int tid = threadIdx.x;
  const int l = tid & 31, w = tid >> 5;
  const int mi = w >> 1, ni = w & 1;   // 4 M-waves x 2 N-waves
  const int ln = l & 15, h = l >> 4;
  const int m0 = blockIdx.x * 64 + mi * 16;

  // A tile register-resident: a[s] = enc[m0+ln][4s + 2h .. +1]
  v2f a[KSTEPS];
  const float* ap = enc + (size_t)(m0 + ln) * Dk + 2 * h;
#pragma unroll
  for (int s = 0; s < KSTEPS; ++s) a[s] = *(const v2f*)(ap + 4 * s);

  // workgroup-relative LDS byte offset of buf (low 32 bits of flat cast)
  const unsigned lds0 = (unsigned)(unsigned long long)(uintptr_t)(void*)buf;

  // async-stage unit u (tile u>>1, K-chunk u&1) into buffer (u&1)
  auto stage = [&](int u) {
    const int nb = (u >> 1) * 32, kc = u & 1, bs = u & 1;
#pragma unroll
    for (int j = 0; j < 4; ++j) {
      int lin = tid + j * 256;                // 1024 float4 slots
      int row = lin >> 5, c4 = lin & 31;      // 32 float4 per row
      const float* g = cb + (size_t)(nb + row) * Dk + kc * 128 + c4 * 4;
      unsigned lo = lds0 + (unsigned)((bs * 4224 + row * 132 + c4 * 4) * 4);
      asm volatile("global_load_async_to_lds_b128 %0, %1, off"
                   :: "v"(lo), "v"((unsigned long long)(uintptr_t)g)
                   : "memory");
    }
  };

  float minv[8];
  int   mini[8];
#pragma unroll
  for (int r = 0; r < 8; ++r) { minv[r] = __builtin_inff(); mini[r] = 0; }

  constexpr int U = (NCB / 32) * 2;  // 512 pipeline units
  stage(0);
  stage(1);

  v8f acc = {};
  for (int u = 0; u < U; ++u) {
    // retire unit u's async loads (unit u+1 may stay in flight: in-order)
    if (u + 1 < U) asm volatile("s_wait_asynccnt 0x4" ::: "memory");
    else           asm volatile("s_wait_asynccnt 0x0" ::: "memory");
    __syncthreads();  // data from all waves' async copies visible

    const int kc = u & 1;
    const float* bp = &buf[(u & 1) * 4224 + (ni * 16 + ln) * 132 + 2 * h];
#pragma unroll
    for (int s = 0; s < 32; ++s) {
      v2f b = *(const v2f*)(bp + 4 * s);
      acc = WMMA_F32(a[kc * 32 + s], b, acc);
    }

    if (kc == 1) {  // full K=256 accumulated for this 32-code tile
      const int nb = (u >> 1) * 32;
      const int idx = nb + ni * 16 + ln;
      const float c2v = c2g[idx];           // L2-resident, one scalar/lane
#pragma unroll
      for (int r = 0; r < 8; ++r) {
        float sc = c2v - 2.0f * acc[r];     // ||c||^2 - 2 e.c
        if (sc < minv[r] || (sc == minv[r] && idx < mini[r])) {
          minv[r] = sc; mini[r] = idx;
        }
      }
      acc = (v8f){};
    }

    __syncthreads();                 // everyone done reading buf[u&1]
    if (u + 2 < U) stage(u + 2);     // refill the buffer just drained
  }

  // reduce (min,idx) across the 16 lanes holding each row's columns
#pragma unroll
  for (int r = 0; r < 8; ++r) {
    float v = minv[r];
    int   id = mini[r];
#pragma unroll
    for (int off = 8; off >= 1; off >>= 1) {
      float ov = __shfl_xor(v, off, 32);
      int   oi = __shfl_xor(id, off, 32);
      if (ov < v || (ov == v && oi < id)) { v = ov; id = oi; }
    }
    if (ln == 0) {
      int row = mi * 16 + r + 8 * h;
      redv[row][ni] = v; redi[row][ni] = id;
    }
  }
  __syncthreads();
  if (tid < 64) {
    float v0 = redv[tid][0], v1 = redv[tid][1];
    int   i0 = redi[tid][0], i1 = redi[tid][1];
    int best = (v1 < v0 || (v1 == v0 && i1 < i0)) ? i1 : i0;
    outidx[blockIdx.x * 64 + tid] = best;
  }
}

// ---------------------------------------------------------------------------
__global__ void vq_zero_kernel(float* p) { *p = 0.0f; }

// gather z_q, straight-through output, loss = 1.25*mean|z_q - z_e|
__global__ __launch_bounds__(256, 1)
void vq_gather_kernel(const float* __restrict__ enc, const float* __restrict__ cb,
                      const int* __restrict__ idxs, float* __restrict__ outz,
                      float* __restrict__ outloss, float* __restrict__ outidxf) {
  __shared__ float wsum[8];
  const int row = blockIdx.x, t = threadIdx.x;
  const int id = idxs[row];
  float z = cb[(size_t)id * Dk + t];
  float e = enc[(size_t)row * Dk + t];
  outz[(size_t)row * Dk + t] = e + (z - e);  // z_q_st
  float ad = fabsf(z - e);
#pragma unroll
  for (int off = 16; off >= 1; off >>= 1) ad += __shfl_xor(ad, off, 32);
  if ((t & 31) == 0) wsum[t >> 5] = ad;
  __syncthreads();
  if (t == 0) {
    float s = 0.f;
#pragma unroll
    for (int i = 0; i < 8; ++i) s += wsum[i];
    atomicAdd(outloss, s * (1.25f / (float)((size_t)MROWS * Dk)));
    outidxf[row] = (float)id;
  }
}

// ---------------------------------------------------------------------------
extern "C" void kernel_launch(void* const* d_in, const int* in_sizes, int n_in,
                              void* d_out, int out_size, void* d_ws, size_t ws_size,
                              hipStream_t stream) {
  const float* z_e = (const float*)d_in[0];  // [16,2048,256]
  const float* cw  = (const float*)d_in[1];  // [8192,256]
  const float* pw  = (const float*)d_in[2];  // [256,256]
  const float* pb  = (const float*)d_in[3];  // [256]

  float* out = (float*)d_out;                      // z_q_st | loss | indices
  float* cb  = (float*)d_ws;                       // [8192*256]
  float* c2  = cb + (size_t)NCB * Dk;              // [8192]
  int*   idxs = (int*)(c2 + NCB);                  // [32768]

  float* loss_p = out + (size_t)MROWS * Dk;
  float* idxf_p = loss_p + 1;

  vq_proj_kernel<<<(NCB / 16) * (Dk / 16) / 4, 128, 0, stream>>>(cw, pw, pb, cb);
  vq_c2_kernel<<<NCB / 8, 256, 0, stream>>>(cb, c2);
  vq_dist_kernel<<<MROWS / 64, 256, 0, stream>>>(z_e, cb, c2, idxs);
  vq_zero_kernel<<<1, 1, 0, stream>>>(loss_p);
  vq_gather_kernel<<<MROWS, 256, 0, stream>>>(z_e, cb, idxs, out, loss_p, idxf_p);
}